// GAT_layer_56238301774617
// MI455X (gfx1250) — compile-verified
//
#include <hip/hip_runtime.h>

#define N_NODES 50000
#define N_EDGES 400000
#define F_IN    128
#define F_OUT   128
#define N_HEADS 2
#define C_TOT   (N_HEADS * F_OUT)   // 256 channels, col = head*128 + o

typedef float v2f __attribute__((ext_vector_type(2)));
typedef float v8f __attribute__((ext_vector_type(8)));

// float atomic-max via integer atomics (works with m initialized to -inf bits)
__device__ __forceinline__ void atomicMaxF(float* addr, float val) {
    if (val >= 0.0f)
        atomicMax((int*)addr, __float_as_int(val));
    else
        atomicMin((unsigned int*)addr, __float_as_uint(val));
}

// ---------------------------------------------------------------------------
// init: m = -inf, s = 0, agg = 0
// ---------------------------------------------------------------------------
__global__ void __launch_bounds__(256) gat_init(float* mmax, float* s, float* agg) {
    size_t i = (size_t)blockIdx.x * 256 + threadIdx.x;
    if (i < (size_t)N_NODES * C_TOT) {
        ((unsigned int*)mmax)[i] = 0xFF800000u;   // -inf
        s[i]   = 0.0f;
        agg[i] = 0.0f;
    }
}

// ---------------------------------------------------------------------------
// h[n, head*128+o] = sum_f x[n,f] * Wf[head,f,o] + bf[head,o]
// 16-node tile per block, 8 waves, each wave does 2 of 16 col-tiles (2 heads x 8)
// ---------------------------------------------------------------------------
__global__ void __launch_bounds__(256) gat_h_gemm(const float* __restrict__ x,
                                                  const float* __restrict__ Wf,
                                                  const float* __restrict__ bf,
                                                  float* __restrict__ h) {
    __shared__ float xt[16 * 132];                // stride 132: bank-conflict-free
    const int base = blockIdx.x * 16;
    const int tid  = threadIdx.x;

    for (int idx = tid; idx < 16 * 128; idx += 256) {
        int row = idx >> 7, col = idx & 127;
        int node = base + row;
        xt[row * 132 + col] = (node < N_NODES) ? x[(size_t)node * F_IN + col] : 0.0f;
    }
    __syncthreads();

    const int wave  = tid >> 5;
    const int lane  = tid & 31;
    const int m     = lane & 15;                  // A row (ISA f32 16x4 layout)
    const int koff0 = (lane >> 4) << 1;           // K sub-offset per lane half

    for (int t = wave; t < 16; t += 8) {
        const int head = t >> 3;
        const int n0   = (t & 7) << 4;
        const int n    = n0 + (lane & 15);
        const float* W = Wf + (size_t)head * F_IN * F_OUT;

        v8f acc = {};
        for (int k = 0; k < F_IN; k += 4) {
            const int ko = k + koff0;
            v2f a = *(const v2f*)&xt[m * 132 + ko];
            v2f b;
            b.x = W[(size_t)ko * F_OUT + n];
            b.y = W[(size_t)(ko + 1) * F_OUT + n];
            acc = __builtin_amdgcn_wmma_f32_16x16x4_f32(false, a, false, b,
                                                        (short)0, acc, false, false);
        }
        const float bias = bf[head * F_OUT + n];
        #pragma unroll
        for (int v = 0; v < 8; ++v) {
            int row  = v + ((lane >> 4) << 3);    // C/D layout: lanes16-31 -> M+8
            int node = base + row;
            if (node < N_NODES)
                h[(size_t)node * C_TOT + head * F_OUT + n] = acc[v] + bias;
        }
    }
}

// ---------------------------------------------------------------------------
// e[E, head*128+o] = LeakyReLU([x[src],x[dst]] . Wa[head] + ba[head])
// fused segment-max (atomicMaxF into mmax[dst])
// ---------------------------------------------------------------------------
__global__ void __launch_bounds__(256) gat_att_gemm(const float* __restrict__ x,
                                                    const int* __restrict__ srcIdx,
                                                    const int* __restrict__ dstIdx,
                                                    const float* __restrict__ Wa,
                                                    const float* __restrict__ ba,
                                                    float* __restrict__ e,
                                                    float* __restrict__ mmax) {
    __shared__ float ct[16 * 260];                // 16 edges x 256 (pad 4)
    __shared__ int   dsts[16];
    const int base = blockIdx.x * 16;
    const int tid  = threadIdx.x;

    if (tid < 16) {
        int edge  = base + tid;
        dsts[tid] = (edge < N_EDGES) ? dstIdx[edge] : 0;
    }
    for (int idx = tid; idx < 16 * 256; idx += 256) {
        int row = idx >> 8, col = idx & 255;
        int edge = base + row;
        float val = 0.0f;
        if (edge < N_EDGES) {
            int node = (col < F_IN) ? srcIdx[edge] : dstIdx[edge];
            val = x[(size_t)node * F_IN + (col & 127)];
        }
        ct[row * 260 + col] = val;
    }
    __syncthreads();

    const int wave  = tid >> 5;
    const int lane  = tid & 31;
    const int m     = lane & 15;
    const int koff0 = (lane >> 4) << 1;

    for (int t = wave; t < 16; t += 8) {
        const int head = t >> 3;
        const int n0   = (t & 7) << 4;
        const int n    = n0 + (lane & 15);
        const float* W = Wa + (size_t)head * (2 * F_IN) * F_OUT;

        v8f acc = {};
        for (int k = 0; k < 2 * F_IN; k += 4) {
            const int ko = k + koff0;
            v2f a = *(const v2f*)&ct[m * 260 + ko];
            v2f b;
            b.x = W[(size_t)ko * F_OUT + n];
            b.y = W[(size_t)(ko + 1) * F_OUT + n];
            acc = __builtin_amdgcn_wmma_f32_16x16x4_f32(false, a, false, b,
                                                        (short)0, acc, false, false);
        }
        const float bias = ba[head * F_OUT + n];
        #pragma unroll
        for (int v = 0; v < 8; ++v) {
            int row  = v + ((lane >> 4) << 3);
            int edge = base + row;
            if (edge < N_EDGES) {
                float val = acc[v] + bias;
                val = (val > 0.0f) ? val : 0.01f * val;          // LeakyReLU
                e[(size_t)edge * C_TOT + head * F_OUT + n] = val;
                atomicMaxF(&mmax[(size_t)dsts[row] * C_TOT + head * F_OUT + n], val);
            }
        }
    }
}

// ---------------------------------------------------------------------------
// z = exp(e - m[dst]) (in place), s[dst] += z
// ---------------------------------------------------------------------------
__global__ void __launch_bounds__(256) gat_exp_sum(float* ez,
                                                   const int* __restrict__ dstIdx,
                                                   const float* __restrict__ mmax,
                                                   float* __restrict__ s) {
    size_t i = (size_t)blockIdx.x * 256 + threadIdx.x;   // grid covers E*256 exactly
    int edge = (int)(i >> 8);
    int c    = (int)(i & 255);
    int d    = dstIdx[edge];
    float z  = expf(ez[i] - mmax[(size_t)d * C_TOT + c]);
    ez[i] = z;
    atomicAdd(&s[(size_t)d * C_TOT + c], z);
}

// ---------------------------------------------------------------------------
// agg[dst] += h[src] * z / (s[dst] + 1e-9)
// ---------------------------------------------------------------------------
__global__ void __launch_bounds__(256) gat_aggregate(const float* __restrict__ z,
                                                     const int* __restrict__ srcIdx,
                                                     const int* __restrict__ dstIdx,
                                                     const float* __restrict__ s,
                                                     const float* __restrict__ h,
                                                     float* __restrict__ agg) {
    size_t i = (size_t)blockIdx.x * 256 + threadIdx.x;
    int edge = (int)(i >> 8);
    int c    = (int)(i & 255);
    int d    = dstIdx[edge];
    int sr   = srcIdx[edge];
    float alpha = z[i] / (s[(size_t)d * C_TOT + c] + 1e-9f);
    atomicAdd(&agg[(size_t)d * C_TOT + c], h[(size_t)sr * C_TOT + c] * alpha);
}

// ---------------------------------------------------------------------------
// out = sigmoid(agg)
// ---------------------------------------------------------------------------
__global__ void __launch_bounds__(256) gat_sigmoid(const float* __restrict__ agg,
                                                   float* __restrict__ out) {
    size_t i = (size_t)blockIdx.x * 256 + threadIdx.x;
    if (i < (size_t)N_NODES * C_TOT)
        out[i] = 1.0f / (1.0f + expf(-agg[i]));
}

// ---------------------------------------------------------------------------
extern "C" void kernel_launch(void* const* d_in, const int* in_sizes, int n_in,
                              void* d_out, int out_size, void* d_ws, size_t ws_size,
                              hipStream_t stream) {
    (void)in_sizes; (void)n_in; (void)out_size; (void)ws_size;
    const float* x  = (const float*)d_in[0];
    const int*   ei = (const int*)d_in[1];     // [2, E]: row 0 = src, row 1 = dst
    const float* Wa = (const float*)d_in[2];   // [H, 256, 128]
    const float* ba = (const float*)d_in[3];   // [H, 128]
    const float* Wf = (const float*)d_in[4];   // [H, 128, 128]
    const float* bf = (const float*)d_in[5];   // [H, 128]
    const int* srcIdx = ei;
    const int* dstIdx = ei + N_EDGES;
    float* out = (float*)d_out;

    char* ws = (char*)d_ws;
    const size_t nodeBytes = (size_t)N_NODES * C_TOT * sizeof(float);  // 51.2 MB
    float* h    = (float*)(ws);
    float* mmax = (float*)(ws + 1 * nodeBytes);
    float* s    = (float*)(ws + 2 * nodeBytes);
    float* agg  = (float*)(ws + 3 * nodeBytes);
    float* e    = (float*)(ws + 4 * nodeBytes);                        // 409.6 MB

    const int nodeElemBlocks = (int)(((size_t)N_NODES * C_TOT + 255) / 256);
    const int edgeElemBlocks = (int)(((size_t)N_EDGES * C_TOT) / 256); // exact

    gat_init<<<nodeElemBlocks, 256, 0, stream>>>(mmax, s, agg);
    gat_h_gemm<<<(N_NODES + 15) / 16, 256, 0, stream>>>(x, Wf, bf, h);
    gat_att_gemm<<<(N_EDGES + 15) / 16, 256, 0, stream>>>(x, srcIdx, dstIdx, Wa, ba, e, mmax);
    gat_exp_sum<<<edgeElemBlocks, 256, 0, stream>>>(e, dstIdx, mmax, s);
    gat_aggregate<<<edgeElemBlocks, 256, 0, stream>>>(e, srcIdx, dstIdx, s, h, agg);
    gat_sigmoid<<<nodeElemBlocks, 256, 0, stream>>>(agg, out);
}